// SAGE_17463337025713
// MI455X (gfx1250) — compile-verified
//
#include <hip/hip_runtime.h>
#include <hip/hip_bf16.h>

typedef __attribute__((ext_vector_type(2))) float v2f;
typedef __attribute__((ext_vector_type(8))) float v8f;

// ---------------------------------------------------------------------------
// zero fill
// ---------------------------------------------------------------------------
__global__ void zero_kernel(float* __restrict__ p, size_t n) {
    size_t i = (size_t)blockIdx.x * blockDim.x + threadIdx.x;
    if (i < n) p[i] = 0.0f;
}

// ---------------------------------------------------------------------------
// degree: deg[dst] += 1 per edge
// ---------------------------------------------------------------------------
__global__ void degree_kernel(const long long* __restrict__ dst,
                              float* __restrict__ deg, int E) {
    int e = blockIdx.x * blockDim.x + threadIdx.x;
    if (e < E) atomicAdd(&deg[(int)dst[e]], 1.0f);
}

// deg[i] = 1 / max(deg[i], 1)   (in place)
__global__ void invdeg_kernel(float* __restrict__ deg, int n) {
    int i = blockIdx.x * blockDim.x + threadIdx.x;
    if (i < n) deg[i] = 1.0f / fmaxf(deg[i], 1.0f);
}

// ---------------------------------------------------------------------------
// scatter-add: one wave32 per edge, lane handles 4 consecutive features
// agg[dst, :] += h[src, :]   (D = 128)
// ---------------------------------------------------------------------------
__global__ __launch_bounds__(256)
void scatter_add_kernel(const float* __restrict__ h,
                        const long long* __restrict__ srcArr,
                        const long long* __restrict__ dstArr,
                        float* __restrict__ agg, int E) {
    int t = blockIdx.x * blockDim.x + threadIdx.x;
    int e = t >> 5;
    int lane = t & 31;
    if (e >= E) return;
    int s = (int)srcArr[e];
    int d = (int)dstArr[e];
    const float4 v = *(const float4*)(h + (size_t)s * 128 + lane * 4);
    float* dp = agg + (size_t)d * 128 + lane * 4;
    atomicAdd(dp + 0, v.x);
    atomicAdd(dp + 1, v.y);
    atomicAdd(dp + 2, v.z);
    atomicAdd(dp + 3, v.w);
}

// ---------------------------------------------------------------------------
// Fused SAGE layer GEMM via V_WMMA_F32_16X16X4_F32:
//   out[64 x DO] = [mean | h] (64 x 256) @ [Wl.T ; Wr.T] (256 x DO) + bl
// Block = 64 nodes (4 M-tiles of 16), DO/16 waves; each B-fragment load is
// reused across 4 WMMAs (4x less weight traffic from L2).
// LDS row stride 260 floats -> conflict-free column reads (260 % 64 == 4).
// Split K loop (Wl phase / Wr phase) keeps B base pointers static.
// ---------------------------------------------------------------------------
template <int DO, bool RELU>
__global__ __launch_bounds__(DO * 2)
void sage_gemm_kernel(const float* __restrict__ agg,
                      const float* __restrict__ hself,
                      const float* __restrict__ invdeg,
                      const float* __restrict__ Wl,
                      const float* __restrict__ bl,
                      const float* __restrict__ Wr,
                      float* __restrict__ out, int N) {
    constexpr int MT    = 4;     // M tiles per block (64 nodes)
    constexpr int LDSRW = 260;
    __shared__ float sA[64 * LDSRW];

    const int tid   = threadIdx.x;
    const int node0 = blockIdx.x * 64;

    // cooperative stage: 64 rows x (32 mean-float4 + 32 h-float4)
    for (int idx = tid; idx < 64 * 64; idx += blockDim.x) {
        int row  = idx >> 6;
        int q    = idx & 63;
        int node = node0 + row;
        float4 v = make_float4(0.f, 0.f, 0.f, 0.f);
        if (q < 32) {
            if (node < N) {
                v = *(const float4*)(agg + (size_t)node * 128 + q * 4);
                float sc = invdeg[node];
                v.x *= sc; v.y *= sc; v.z *= sc; v.w *= sc;
            }
            *(float4*)(&sA[row * LDSRW + q * 4]) = v;
        } else {
            if (node < N) v = *(const float4*)(hself + (size_t)node * 128 + (q - 32) * 4);
            *(float4*)(&sA[row * LDSRW + 128 + (q - 32) * 4]) = v;
        }
    }
    __syncthreads();

    const int lane  = tid & 31;
    const int wave  = tid >> 5;
    const int half  = lane >> 4;   // K sub-chunk select (ISA 32-bit A/B layout)
    const int idx16 = lane & 15;   // A: row M   B: col N
    const int o     = wave * 16 + idx16;

    const float* wlrow = Wl + (size_t)o * 128;
    const float* wrrow = Wr + (size_t)o * 128;

    v8f acc[MT];
#pragma unroll
    for (int t = 0; t < MT; ++t)
        acc[t] = (v8f){0.f, 0.f, 0.f, 0.f, 0.f, 0.f, 0.f, 0.f};

    // ---- phase 1: mean part vs Wl.T (K = 0..127)
#pragma unroll 4
    for (int kk = 0; kk < 128; kk += 4) {
        const int k0 = kk + half * 2;
        const float2 bw = *(const float2*)(wlrow + k0);
        v2f b; b.x = bw.x; b.y = bw.y;
#pragma unroll
        for (int t = 0; t < MT; ++t) {
            const float* arow = &sA[(t * 16 + idx16) * LDSRW];
            v2f a; a.x = arow[k0]; a.y = arow[k0 + 1];
            acc[t] = __builtin_amdgcn_wmma_f32_16x16x4_f32(
                false, a, false, b, (short)0, acc[t], false, false);
        }
    }
    // ---- phase 2: self part vs Wr.T (K = 128..255)
#pragma unroll 4
    for (int kk = 0; kk < 128; kk += 4) {
        const int k0 = kk + half * 2;
        const float2 bw = *(const float2*)(wrrow + k0);
        v2f b; b.x = bw.x; b.y = bw.y;
#pragma unroll
        for (int t = 0; t < MT; ++t) {
            const float* arow = &sA[(t * 16 + idx16) * LDSRW + 128];
            v2f a; a.x = arow[k0]; a.y = arow[k0 + 1];
            acc[t] = __builtin_amdgcn_wmma_f32_16x16x4_f32(
                false, a, false, b, (short)0, acc[t], false, false);
        }
    }

    const float bias = bl[o];
    if (node0 + 64 <= N) {
        // fast path: full tile, unguarded stores
#pragma unroll
        for (int t = 0; t < MT; ++t) {
#pragma unroll
            for (int r = 0; r < 8; ++r) {
                int node = node0 + t * 16 + r + half * 8;  // C/D: reg r, half -> M+8
                float v = acc[t][r] + bias;
                if (RELU) v = fmaxf(v, 0.0f);
                out[(size_t)node * DO + o] = v;
            }
        }
    } else {
#pragma unroll
        for (int t = 0; t < MT; ++t) {
#pragma unroll
            for (int r = 0; r < 8; ++r) {
                int node = node0 + t * 16 + r + half * 8;
                if (node < N) {
                    float v = acc[t][r] + bias;
                    if (RELU) v = fmaxf(v, 0.0f);
                    out[(size_t)node * DO + o] = v;
                }
            }
        }
    }
}

// ---------------------------------------------------------------------------
// log_softmax over 64 classes: one wave per node, 2 values per lane
// ---------------------------------------------------------------------------
__global__ __launch_bounds__(256)
void log_softmax64_kernel(const float* __restrict__ in,
                          float* __restrict__ out, int N) {
    int t = blockIdx.x * blockDim.x + threadIdx.x;
    int node = t >> 5;
    int lane = t & 31;
    if (node >= N) return;
    const float* row = in + (size_t)node * 64;
    float a = row[lane];
    float b = row[lane + 32];
    float mx = fmaxf(a, b);
#pragma unroll
    for (int off = 16; off > 0; off >>= 1)
        mx = fmaxf(mx, __shfl_xor(mx, off, 32));
    float s = expf(a - mx) + expf(b - mx);
#pragma unroll
    for (int off = 16; off > 0; off >>= 1)
        s += __shfl_xor(s, off, 32);
    float lse = logf(s) + mx;
    out[(size_t)node * 64 + lane]      = a - lse;
    out[(size_t)node * 64 + lane + 32] = b - lse;
}

// ---------------------------------------------------------------------------
// launch
// ---------------------------------------------------------------------------
extern "C" void kernel_launch(void* const* d_in, const int* in_sizes, int n_in,
                              void* d_out, int out_size, void* d_ws, size_t ws_size,
                              hipStream_t stream) {
    const float*     x   = (const float*)d_in[0];
    const long long* ei  = (const long long*)d_in[1];   // int64 per reference
    const float*     Wl0 = (const float*)d_in[2];
    const float*     bl0 = (const float*)d_in[3];
    const float*     Wr0 = (const float*)d_in[4];
    const float*     Wl1 = (const float*)d_in[5];
    const float*     bl1 = (const float*)d_in[6];
    const float*     Wr1 = (const float*)d_in[7];
    const float*     Wl2 = (const float*)d_in[8];
    const float*     bl2 = (const float*)d_in[9];
    const float*     Wr2 = (const float*)d_in[10];
    float*           out = (float*)d_out;

    const int N = in_sizes[0] / 128;
    const int E = in_sizes[1] / 2;
    const long long* dst = ei;       // edge_index[0]
    const long long* src = ei + E;   // edge_index[1]

    // workspace layout (floats): deg[N] | agg[N*128] | h1[N*128] | h2[N*128]
    float* deg = (float*)d_ws;
    float* agg = deg + N;
    float* h1  = agg + (size_t)N * 128;
    float* h2  = h1 + (size_t)N * 128;

    const size_t nAgg = (size_t)N * 128;
    const int    ZB   = 256;
    auto zgrid = [](size_t n) { return (int)((n + 255) / 256); };
    const int scatterGrid = (int)(((size_t)E * 32 + 255) / 256);
    const int gemmGrid    = (N + 63) / 64;
    const int waveGrid    = (int)(((size_t)N * 32 + 255) / 256);

    // degree (once)
    zero_kernel<<<zgrid(N), ZB, 0, stream>>>(deg, (size_t)N);
    degree_kernel<<<(E + 255) / 256, 256, 0, stream>>>(dst, deg, E);
    invdeg_kernel<<<(N + 255) / 256, 256, 0, stream>>>(deg, N);

    // ---- layer 0: x -> h1 (relu)
    zero_kernel<<<zgrid(nAgg), ZB, 0, stream>>>(agg, nAgg);
    scatter_add_kernel<<<scatterGrid, 256, 0, stream>>>(x, src, dst, agg, E);
    sage_gemm_kernel<128, true><<<gemmGrid, 256, 0, stream>>>(
        agg, x, deg, Wl0, bl0, Wr0, h1, N);

    // ---- layer 1: h1 -> h2 (relu)
    zero_kernel<<<zgrid(nAgg), ZB, 0, stream>>>(agg, nAgg);
    scatter_add_kernel<<<scatterGrid, 256, 0, stream>>>(h1, src, dst, agg, E);
    sage_gemm_kernel<128, true><<<gemmGrid, 256, 0, stream>>>(
        agg, h1, deg, Wl1, bl1, Wr1, h2, N);

    // ---- layer 2: h2 -> logits (reuse h1 buffer), no relu
    zero_kernel<<<zgrid(nAgg), ZB, 0, stream>>>(agg, nAgg);
    scatter_add_kernel<<<scatterGrid, 256, 0, stream>>>(h2, src, dst, agg, E);
    sage_gemm_kernel<64, false><<<gemmGrid, 128, 0, stream>>>(
        agg, h2, deg, Wl2, bl2, Wr2, h1, N);

    // ---- log_softmax
    log_softmax64_kernel<<<waveGrid, 256, 0, stream>>>(h1, out, N);
}